// PatchCore_12799002542630
// MI455X (gfx1250) — compile-verified
//
#include <hip/hip_runtime.h>
#include <hip/hip_bf16.h>
#include <math.h>

typedef __attribute__((ext_vector_type(2))) float v2f;
typedef __attribute__((ext_vector_type(8))) float v8f;

#define Q_TOTAL 3136
#define DIMS    128
#define M_TOTAL 32768
#define NCHUNK  8
#define MCHUNK  4096
#define QB      32      // queries per workgroup (2 tiles of 16)
#define NT      32      // memory columns per inner iteration (2 tiles of 16)
#define LDSS    132     // padded LDS row stride (floats); 132 % 64 = 4 -> conflict-free
#define KNN     9
#define FLT_BIG 3.402823466e38f

__device__ __forceinline__ void topk_insert(float (&t)[KNN], float v) {
    if (v < t[KNN - 1]) {
        t[KNN - 1] = v;
#pragma unroll
        for (int i = KNN - 1; i > 0; --i) {
            if (t[i] < t[i - 1]) { float tmp = t[i - 1]; t[i - 1] = t[i]; t[i] = tmp; }
        }
    }
}

// --- K1: 3x3 avg pool (count_include_pad, /9) + per-query squared norm -----
__global__ void __launch_bounds__(128)
avgpool_kernel(const float* __restrict__ feat, float* __restrict__ pooled,
               float* __restrict__ qsq) {
    __shared__ float red[128];
    const int q = blockIdx.x;            // 0..3135
    const int d = threadIdx.x;           // 0..127
    const int b = q / 784;
    const int rem = q - b * 784;
    const int h = rem / 28;
    const int w = rem - h * 28;
    float s = 0.f;
    for (int dy = -1; dy <= 1; ++dy) {
        int hh = h + dy;
        if (hh < 0 || hh > 27) continue;
        for (int dx = -1; dx <= 1; ++dx) {
            int ww = w + dx;
            if (ww < 0 || ww > 27) continue;
            s += feat[(((size_t)(b * 28 + hh)) * 28 + ww) * DIMS + d];
        }
    }
    s *= (1.0f / 9.0f);
    pooled[(size_t)q * DIMS + d] = s;
    red[d] = s * s;
    __syncthreads();
    for (int st = 64; st > 0; st >>= 1) {
        if (d < st) red[d] += red[d + st];
        __syncthreads();
    }
    if (d == 0) qsq[q] = red[0];
}

// --- K2: per-memory-row squared norm (one wave32 per row, float4 lanes) ----
__global__ void __launch_bounds__(256)
msq_kernel(const float* __restrict__ mem, float* __restrict__ msq) {
    const int lane = threadIdx.x & 31;
    const int wave = threadIdx.x >> 5;
    const int row  = blockIdx.x * 8 + wave;
    const float4 v = ((const float4*)mem)[(size_t)row * (DIMS / 4) + lane];
    float s = v.x * v.x + v.y * v.y + v.z * v.z + v.w * v.w;
    for (int off = 16; off > 0; off >>= 1) s += __shfl_xor(s, off, 32);
    if (lane == 0) msq[row] = s;
}

// --- K3: fused fp32-WMMA GEMM + running top-9 of (msq - 2*q.m) per query ---
// grid (Q_TOTAL/QB, NCHUNK), block 128 (4 waves). Wave w owns output tile
// (qtile = w>>1, ctile = w&1) of each 32x32 iteration slab.
__global__ void __launch_bounds__(128)
gemm_topk_kernel(const float* __restrict__ pooled, const float* __restrict__ mem,
                 const float* __restrict__ msq, float* __restrict__ topbuf) {
    __shared__ float As[QB * LDSS];
    __shared__ float Bs[NT * LDSS];
    __shared__ float Dt[4][16 * 17];     // per-wave private distance staging

    const int tid  = threadIdx.x;
    const int lane = tid & 31;
    const int wave = tid >> 5;
    const int qt   = wave >> 1;          // 0..1
    const int ct   = wave & 1;           // 0..1
    const int qbase  = blockIdx.x * QB;
    const int chunk  = blockIdx.y;
    const int mstart = chunk * MCHUNK;

    // stage A (32 queries x 128) once
    for (int i = tid; i < QB * (DIMS / 4); i += 128) {
        int r  = i >> 5;
        int c4 = (i & 31) << 2;
        const float4 v = ((const float4*)pooled)[(size_t)(qbase + r) * (DIMS / 4) + (i & 31)];
        *(float4*)&As[r * LDSS + c4] = v;
    }

    float t9[KNN];
#pragma unroll
    for (int i = 0; i < KNN; ++i) t9[i] = FLT_BIG;

    const int arow    = qt * 16 + (lane & 15);
    const int brow    = ct * 16 + (lane & 15);
    const int koff    = (lane >> 4) << 1;   // 0 or 2 (K sub-offset per A/B layout)
    const int qloc_hi = (lane >> 4) << 3;   // 0 or 8 (C-layout row offset / scan half)

    for (int mt = 0; mt < MCHUNK; mt += NT) {
        __syncthreads();
        for (int i = tid; i < NT * (DIMS / 4); i += 128) {
            int r  = i >> 5;
            int c4 = (i & 31) << 2;
            const float4 v = ((const float4*)mem)[(size_t)(mstart + mt + r) * (DIMS / 4) + (i & 31)];
            *(float4*)&Bs[r * LDSS + c4] = v;
        }
        __syncthreads();

        v8f acc = {};
#pragma unroll
        for (int k0 = 0; k0 < DIMS; k0 += 4) {
            v2f a = *(const v2f*)&As[arow * LDSS + k0 + koff];
            v2f b = *(const v2f*)&Bs[brow * LDSS + k0 + koff];
            acc = __builtin_amdgcn_wmma_f32_16x16x4_f32(
                false, a, false, b, (short)0, acc, false, false);
        }

        // e = ||m||^2 - 2*(q.m)  (||q||^2 is constant per query -> order-invariant)
        const float em = msq[mstart + mt + ct * 16 + (lane & 15)];
#pragma unroll
        for (int r = 0; r < 8; ++r) {
            Dt[wave][(r + qloc_hi) * 17 + (lane & 15)] = em - 2.0f * acc[r];
        }
        // lane owns query (lane&15); low/high half-wave split the 16 candidates
        const int ql = lane & 15;
#pragma unroll
        for (int j = 0; j < 8; ++j) {
            topk_insert(t9, Dt[wave][ql * 17 + qloc_hi + j]);
        }
    }

    // merge the two half-wave candidate lists for each query
#pragma unroll
    for (int i = 0; i < KNN; ++i) {
        float o = __shfl(t9[i], (lane & 15) + 16, 32);
        if (lane < 16) topk_insert(t9, o);
    }
    if (lane < 16) {
        const int q = qbase + qt * 16 + lane;
        float* dst = &topbuf[((size_t)q * NCHUNK + chunk) * KNN];
#pragma unroll
        for (int i = 0; i < KNN; ++i) dst[i] = t9[i];
    }
}

// --- K4: merge 8 partial top-9 lists, sqrt + softmax reweight -> patch score
__global__ void __launch_bounds__(256)
merge_score_kernel(const float* __restrict__ topbuf, const float* __restrict__ qsq,
                   float* __restrict__ ps, float* __restrict__ out) {
    const int q = blockIdx.x * 256 + threadIdx.x;
    if (q < 4) ((int*)out)[q] = 0;   // init per-image score slots for atomicMax
    if (q >= Q_TOTAL) return;
    float t9[KNN];
#pragma unroll
    for (int i = 0; i < KNN; ++i) t9[i] = FLT_BIG;
    for (int c = 0; c < NCHUNK; ++c) {
        const float* src = &topbuf[((size_t)q * NCHUNK + c) * KNN];
        for (int i = 0; i < KNN; ++i) topk_insert(t9, src[i]);
    }
    const float qq = qsq[q];
    float dist[KNN];
#pragma unroll
    for (int i = 0; i < KNN; ++i) dist[i] = sqrtf(fmaxf(t9[i] + qq, 1e-12f));
    // t9 ascending => dist[0] is the nearest => max logit of softmax(-dist)
    float wsum = 0.f, acc = 0.f;
#pragma unroll
    for (int i = 0; i < KNN; ++i) {
        float w = __expf(dist[0] - dist[i]);
        wsum += w;
        acc  += dist[i] * w;
    }
    ps[q] = acc / wsum;
}

// --- K5: bilinear 28->224 (half-pixel centers) + per-image max -------------
__global__ void __launch_bounds__(256)
upsample_max_kernel(const float* __restrict__ ps, float* __restrict__ out) {
    __shared__ float red[256];
    const int idx = blockIdx.x * 256 + threadIdx.x;      // < 4*224*224
    const int b   = idx / (224 * 224);
    const int rem = idx - b * (224 * 224);
    const int y   = rem / 224;
    const int x   = rem - y * 224;
    float sy = (y + 0.5f) * 0.125f - 0.5f;
    float sx = (x + 0.5f) * 0.125f - 0.5f;
    float y0f = floorf(sy), x0f = floorf(sx);
    int y0 = (int)y0f, x0 = (int)x0f;
    float fy = sy - y0f, fx = sx - x0f;
    int y1 = min(y0 + 1, 27); y0 = max(y0, 0);
    int x1 = min(x0 + 1, 27); x0 = max(x0, 0);
    const float* p = &ps[b * 784];
    float v00 = p[y0 * 28 + x0], v01 = p[y0 * 28 + x1];
    float v10 = p[y1 * 28 + x0], v11 = p[y1 * 28 + x1];
    float v0 = v00 + (v01 - v00) * fx;
    float v1 = v10 + (v11 - v10) * fx;
    float v  = v0 + (v1 - v0) * fy;
    out[4 + idx] = v;
    red[threadIdx.x] = v;
    __syncthreads();
    for (int st = 128; st > 0; st >>= 1) {
        if (threadIdx.x < st)
            red[threadIdx.x] = fmaxf(red[threadIdx.x], red[threadIdx.x + st]);
        __syncthreads();
    }
    // scores are strictly positive -> int-bit order == float order
    if (threadIdx.x == 0) atomicMax((int*)out + b, __float_as_int(red[0]));
}

extern "C" void kernel_launch(void* const* d_in, const int* in_sizes, int n_in,
                              void* d_out, int out_size, void* d_ws, size_t ws_size,
                              hipStream_t stream) {
    (void)in_sizes; (void)n_in; (void)out_size; (void)ws_size;
    const float* feat = (const float*)d_in[0];   // (4,28,28,128) fp32
    const float* mem  = (const float*)d_in[1];   // (32768,128) fp32
    // d_in[2] = image_size (224), fixed

    float* ws     = (float*)d_ws;                // ~2.7 MB of fp32 scratch
    float* pooled = ws;                                          // 3136*128
    float* qsq    = pooled + (size_t)Q_TOTAL * DIMS;             // 3136
    float* msq    = qsq + Q_TOTAL;                               // 32768
    float* topbuf = msq + M_TOTAL;                               // 3136*8*9
    float* ps     = topbuf + (size_t)Q_TOTAL * NCHUNK * KNN;     // 3136
    float* out    = (float*)d_out;               // [0..3] scores, [4..] maps

    avgpool_kernel<<<Q_TOTAL, 128, 0, stream>>>(feat, pooled, qsq);
    msq_kernel<<<M_TOTAL / 8, 256, 0, stream>>>(mem, msq);
    dim3 g3(Q_TOTAL / QB, NCHUNK);
    gemm_topk_kernel<<<g3, 128, 0, stream>>>(pooled, mem, msq, topbuf);
    merge_score_kernel<<<(Q_TOTAL + 255) / 256, 256, 0, stream>>>(topbuf, qsq, ps, out);
    upsample_max_kernel<<<784, 256, 0, stream>>>(ps, out);
}